// LocalFeatureSamplerV10_26508538151667
// MI455X (gfx1250) — compile-verified
//
#include <hip/hip_runtime.h>

#define BB  32
#define NN  16384
#define CC  1024
#define KK  32
#define HID 512

typedef float v2f __attribute__((ext_vector_type(2)));
typedef float v8f __attribute__((ext_vector_type(8)));

// ---------------------------------------------------------------------------
// KNN: one block per (query q, batch b). Two 8192-point chunks; per chunk do
// 32 iterative argmin reductions in LDS, then merge the 64 candidates.
// ---------------------------------------------------------------------------
__global__ void knn_kernel(const float* __restrict__ xyz,
                           const float* __restrict__ jq,
                           const float* __restrict__ dq,
                           int* __restrict__ idx_out) {
  const int q   = blockIdx.x / BB;
  const int b   = blockIdx.x % BB;
  const int tid = threadIdx.x;

  const float* qp = (q == 0 ? jq : dq) + b * 3;
  const float qx = qp[0], qy = qp[1], qz = qp[2];
  const float* base = xyz + (size_t)b * NN * 3;

  __shared__ float s_d2[NN / 2];
  __shared__ float s_red[256];
  __shared__ int   s_redi[256];
  __shared__ float s_cd[2 * KK];
  __shared__ int   s_ci[2 * KK];

  for (int chunk = 0; chunk < 2; ++chunk) {
    const int off = chunk * (NN / 2);
    for (int j = tid; j < NN / 2; j += 256) {
      const float* p = base + (size_t)(off + j) * 3;
      const float dx = p[0] - qx, dy = p[1] - qy, dz = p[2] - qz;
      s_d2[j] = dx * dx + dy * dy + dz * dz;
    }
    __syncthreads();

    for (int it = 0; it < KK; ++it) {
      float best = 3.4e38f; int bi = 0;
      for (int j = tid; j < NN / 2; j += 256) {
        const float v = s_d2[j];
        if (v < best) { best = v; bi = j; }
      }
      s_red[tid] = best; s_redi[tid] = bi;
      __syncthreads();
      for (int s = 128; s > 0; s >>= 1) {
        if (tid < s) {
          if (s_red[tid + s] < s_red[tid]) {
            s_red[tid]  = s_red[tid + s];
            s_redi[tid] = s_redi[tid + s];
          }
        }
        __syncthreads();
      }
      if (tid == 0) {
        const int w = s_redi[0];
        s_cd[chunk * KK + it] = s_red[0];
        s_ci[chunk * KK + it] = w + off;
        s_d2[w] = 3.0e38f;                 // remove winner
      }
      __syncthreads();
    }
    __syncthreads();
  }

  if (tid == 0) {                           // merge 64 -> 32 (tiny)
    int* dst = idx_out + (size_t)(q * BB + b) * KK;
    for (int it = 0; it < KK; ++it) {
      float best = 3.4e38f; int bj = 0;
      for (int j = 0; j < 2 * KK; ++j)
        if (s_cd[j] < best) { best = s_cd[j]; bj = j; }
      dst[it] = s_ci[bj];
      s_cd[bj] = 3.4e38f;
    }
  }
}

// ---------------------------------------------------------------------------
// Gather K neighbor columns of point_features [B, C, N] and max-pool over K.
// One block per (q, b); threads stride over C.
// ---------------------------------------------------------------------------
__global__ void gather_max_kernel(const float* __restrict__ pf,
                                  const int* __restrict__ idx,
                                  float* __restrict__ pooled) {
  const int q   = blockIdx.x / BB;
  const int b   = blockIdx.x % BB;
  const int tid = threadIdx.x;

  __shared__ int s_idx[KK];
  if (tid < KK) s_idx[tid] = idx[(size_t)(q * BB + b) * KK + tid];
  __syncthreads();

  const float* basef = pf + (size_t)b * CC * NN;
  float* dst = pooled + (size_t)(q * BB + b) * CC;
  for (int c = tid; c < CC; c += 256) {
    const float* row = basef + (size_t)c * NN;
    float m = -3.4e38f;
#pragma unroll
    for (int k = 0; k < KK; ++k) m = fmaxf(m, row[s_idx[k]]);
    dst[c] = m;
  }
}

// ---------------------------------------------------------------------------
// Fused GEMM + bias (+ReLU) via V_WMMA_F32_16X16X4_F32 (exact fp32).
// One 16x16 output tile per wave; M=32 -> 2 tile rows, N=512 -> 32 tile cols.
// K stepped by 8 with TWO independent accumulator chains (acc0/acc1) so the
// WMMA D->C dependency chain is split and loads overlap matrix issue.
// Base pointers hoisted so the inner loop uses constant-stride bumps instead
// of per-iteration 64-bit address recomputation.
// A 16x4 frag: lanes 0-15 -> M=lane, VGPR0/1 = K0/K1; lanes 16-31 = K2/K3.
// B 4x16 frag mirrored. D: VGPR v = row (v + 8*laneHalf), col = lane&15.
// No divergence -> EXEC all ones (WMMA requirement).
// ---------------------------------------------------------------------------
__global__ void gemm_wmma_kernel(const float* __restrict__ A, int lda,
                                 const float* __restrict__ W, int ldw,
                                 const float* __restrict__ bias,
                                 float* __restrict__ Out, int ldo,
                                 int Ktot, int ntiles, int relu) {
  const int tid  = threadIdx.x;
  const int lane = tid & 31;
  const int wid  = (blockIdx.x * blockDim.x + tid) >> 5;
  const int tm   = wid / ntiles;
  const int tn   = wid % ntiles;
  const int half = lane >> 4;
  const int l16  = lane & 15;
  const int row  = tm * 16 + l16;
  const int col  = tn * 16 + l16;

  // Loop-invariant bases: this lane's A row (offset to its K pair) and W col.
  const float* Ap = A + (size_t)row * lda + 2 * half;
  const float* Wp = W + (size_t)(2 * half) * ldw + col;

  v8f acc0 = {};
  v8f acc1 = {};
  for (int k0 = 0; k0 < Ktot; k0 += 8) {
    v2f a0, b0, a1, b1;
    a0.x = Ap[0];
    a0.y = Ap[1];
    b0.x = Wp[0];
    b0.y = Wp[ldw];
    a1.x = Ap[4];
    a1.y = Ap[5];
    b1.x = Wp[4 * ldw];
    b1.y = Wp[5 * ldw];
    acc0 = __builtin_amdgcn_wmma_f32_16x16x4_f32(
        false, a0, false, b0, (short)0, acc0, false, false);
    acc1 = __builtin_amdgcn_wmma_f32_16x16x4_f32(
        false, a1, false, b1, (short)0, acc1, false, false);
    Ap += 8;
    Wp += (size_t)8 * ldw;
  }

  const float bv = bias[col];
#pragma unroll
  for (int v = 0; v < 8; ++v) {
    float r = acc0[v] + acc1[v] + bv;
    if (relu) r = fmaxf(r, 0.0f);
    Out[(size_t)(tm * 16 + half * 8 + v) * ldo + col] = r;
  }
}

// ---------------------------------------------------------------------------
extern "C" void kernel_launch(void* const* d_in, const int* in_sizes, int n_in,
                              void* d_out, int out_size, void* d_ws, size_t ws_size,
                              hipStream_t stream) {
  const float* xyz = (const float*)d_in[0];
  const float* pf  = (const float*)d_in[1];
  const float* jq  = (const float*)d_in[2];
  const float* dq  = (const float*)d_in[3];
  const float* jw1 = (const float*)d_in[4];
  const float* jb1 = (const float*)d_in[5];
  const float* jw2 = (const float*)d_in[6];
  const float* jb2 = (const float*)d_in[7];
  const float* dw1 = (const float*)d_in[8];
  const float* db1 = (const float*)d_in[9];
  const float* dw2 = (const float*)d_in[10];
  const float* db2 = (const float*)d_in[11];
  const float* fw1 = (const float*)d_in[12];
  const float* fb1 = (const float*)d_in[13];
  const float* fw2 = (const float*)d_in[14];
  const float* fb2 = (const float*)d_in[15];
  float* out = (float*)d_out;

  char*  ws     = (char*)d_ws;
  int*   idx    = (int*)ws;                                         // [2,B,K] int
  float* pooled = (float*)(ws + (size_t)2 * BB * KK * sizeof(int)); // [2,B,C]
  float* h      = pooled + (size_t)2 * BB * CC;                     // [2,B,512]
  float* comb   = h + (size_t)2 * BB * HID;                         // [B,1024]
  float* g      = comb + (size_t)BB * 2 * HID;                      // [B,512]

  const int NT = HID / 16;   // 32 tile cols; 2 tile rows -> 64 waves -> 8 blocks

  knn_kernel       <<<2 * BB, 256, 0, stream>>>(xyz, jq, dq, idx);
  gather_max_kernel<<<2 * BB, 256, 0, stream>>>(pf, idx, pooled);

  // branch layer 1 (K=1024, ReLU)
  gemm_wmma_kernel<<<8, 256, 0, stream>>>(pooled,            CC, jw1, HID, jb1,
                                          h,            HID, CC, NT, 1);
  gemm_wmma_kernel<<<8, 256, 0, stream>>>(pooled + BB * CC,  CC, dw1, HID, db1,
                                          h + BB * HID, HID, CC, NT, 1);
  // branch layer 2 (K=512) -> write directly into concat halves (ldo = 1024)
  gemm_wmma_kernel<<<8, 256, 0, stream>>>(h,            HID, jw2, HID, jb2,
                                          comb,       2 * HID, HID, NT, 0);
  gemm_wmma_kernel<<<8, 256, 0, stream>>>(h + BB * HID, HID, dw2, HID, db2,
                                          comb + HID, 2 * HID, HID, NT, 0);
  // fusion MLP
  gemm_wmma_kernel<<<8, 256, 0, stream>>>(comb, 2 * HID, fw1, HID, fb1,
                                          g,   HID, 2 * HID, NT, 1);
  gemm_wmma_kernel<<<8, 256, 0, stream>>>(g,        HID, fw2, HID, fb2,
                                          out, HID,     HID, NT, 0);
}